// VectorQuantizer_65180423685706
// MI455X (gfx1250) — compile-verified
//
#include <hip/hip_runtime.h>
#include <hip/hip_bf16.h>
#include <math.h>

#define N_ROWS   131072
#define ZDIM     256
#define NCODES   2048
#define CHUNK    64             // codes per LDS buffer (double-buffered)
#define NCHUNK   (NCODES / CHUNK)
#define LDS_STRIDE 260          // padded row stride (floats) -> conflict-free B reads
#define ROWS_PER_WAVE 16
#define WAVES_PER_BLOCK 8
#define ROWS_PER_BLOCK (ROWS_PER_WAVE * WAVES_PER_BLOCK)   // 128

typedef float v2f __attribute__((ext_vector_type(2)));
typedef float v8f __attribute__((ext_vector_type(8)));

// ws layout (float words):
//   [0    .. 2047]  codebook squared norms
//   [2048 .. 4095]  histogram (int32)
//   [4096]          loss accumulator (float)

__global__ void vq_zero(float* __restrict__ enc, long long n_f2, float* __restrict__ ws) {
    long long tid    = (long long)blockIdx.x * blockDim.x + threadIdx.x;
    long long stride = (long long)gridDim.x * blockDim.x;
    v2f* p = (v2f*)enc;                             // enc base is 8B-aligned
    const v2f z = {0.f, 0.f};
    for (long long i = tid; i < n_f2; i += stride)
        __builtin_nontemporal_store(z, &p[i]);      // streaming zeros: keep L2 for codebook
    if (blockIdx.x == 0) {
        for (int i = threadIdx.x; i < 2049; i += blockDim.x)
            ws[2048 + i] = 0.f;                     // hist (int 0 == 0.0f bits) + lossAcc
    }
}

__global__ void vq_cbnorm(const float* __restrict__ cb, float* __restrict__ norms) {
    int wave = (int)((blockIdx.x * blockDim.x + threadIdx.x) >> 5);
    int lane = threadIdx.x & 31;
    if (wave >= NCODES) return;
    const float4* row = (const float4*)(cb + (long long)wave * ZDIM);
    float4 a = row[lane * 2 + 0];
    float4 b = row[lane * 2 + 1];
    float s = a.x*a.x + a.y*a.y + a.z*a.z + a.w*a.w
            + b.x*b.x + b.y*b.y + b.z*b.z + b.w*b.w;
    #pragma unroll
    for (int off = 16; off; off >>= 1) s += __shfl_xor(s, off, 32);
    if (lane == 0) norms[wave] = s;
}

// Issue this wave's share of an async 64-code chunk copy (ASYNCcnt-tracked,
// lands directly in LDS; overlaps with WMMA compute on the other buffer).
__device__ __forceinline__ void stage_async(const float* __restrict__ cb, int chunk,
                                            float* __restrict__ dstbuf, int tid) {
    const unsigned gbase = (unsigned)(chunk * CHUNK * ZDIM * 4);   // byte offset, < 2 MB
    #pragma unroll
    for (int i = 0; i < (CHUNK * (ZDIM / 4)) / 256; ++i) {         // 16 x b128 per lane
        int g    = tid + 256 * i;          // float4 index within chunk
        int code = g >> 6;                 // 64 float4 per code row
        int k4   = g & 63;
        unsigned lds_addr = (unsigned)(uintptr_t)(&dstbuf[code * LDS_STRIDE + k4 * 4]);
        unsigned goff     = gbase + (unsigned)g * 16u;
        asm volatile("global_load_async_to_lds_b128 %0, %1, %2"
                     :: "v"(lds_addr), "v"(goff), "s"(cb)
                     : "memory");
    }
}

__device__ __forceinline__ void wait_async_all() {
    asm volatile("s_wait_asynccnt 0x0" ::: "memory");
}

__launch_bounds__(256, 1)
__global__ void vq_main(const float* __restrict__ x, const float* __restrict__ cb,
                        const float* __restrict__ norms, int* __restrict__ hist,
                        float* __restrict__ lossAcc,
                        float* __restrict__ quant, float* __restrict__ enc) {
    __shared__ float lds[2][CHUNK * LDS_STRIDE];   // 2 x 66,560 B = 130 KB of 320 KB WGP LDS

    const int lane = threadIdx.x & 31;
    const int wid  = threadIdx.x >> 5;
    const int half = lane >> 4;                    // 0 | 1
    const int l16  = lane & 15;
    const int rowBase = blockIdx.x * ROWS_PER_BLOCK + wid * ROWS_PER_WAVE;

    // kick off the first chunk's async staging before touching A
    stage_async(cb, 0, lds[0], threadIdx.x);

    // ---- A fragments for V_WMMA_F32_16X16X4_F32, all 64 K-steps, kept in VGPRs ----
    // lanes 0-15: rows M=0..15 hold (K=4kk+0, 4kk+1); lanes 16-31: same rows, (K=4kk+2, 4kk+3)
    v2f a[64];
    {
        const float* arow = x + (long long)(rowBase + l16) * ZDIM + 2 * half;
        #pragma unroll
        for (int kk = 0; kk < 64; ++kk)
            a[kk] = *(const v2f*)(arow + kk * 4);
    }

    float minD[8];
    int   minI[8];
    #pragma unroll
    for (int j = 0; j < 8; ++j) { minD[j] = 3.4e38f; minI[j] = 0; }

    for (int chunk = 0; chunk < NCHUNK; ++chunk) {
        const float* buf = lds[chunk & 1];
        wait_async_all();                 // my chunk-i loads (and older) have landed
        __syncthreads();                  // everyone's landed; prev compute on other buf done
        if (chunk + 1 < NCHUNK)           // prefetch next chunk into the other buffer
            stage_async(cb, chunk + 1, lds[(chunk + 1) & 1], threadIdx.x);

        #pragma unroll 1
        for (int t = 0; t < CHUNK / 16; ++t) {
            const int c0 = chunk * CHUNK + t * 16;
            v8f c = {};
            // B fragment: lane = code (N) in l16; K pair selected by half (stride-260 -> no bank conflicts)
            const float* brow = &buf[(t * 16 + l16) * LDS_STRIDE + 2 * half];
            #pragma unroll
            for (int kk = 0; kk < 64; ++kk) {
                v2f b = *(const v2f*)(brow + kk * 4);
                c = __builtin_amdgcn_wmma_f32_16x16x4_f32(
                        false, a[kk], false, b, (short)0, c, false, false);
            }
            const int   codeIdx = c0 + l16;
            const float nrm     = norms[codeIdx];
            #pragma unroll
            for (int j = 0; j < 8; ++j) {          // VGPR j -> row M = j + 8*half, col N = l16
                float d = nrm - 2.0f * c[j];
                if (d < minD[j]) { minD[j] = d; minI[j] = codeIdx; }
            }
        }
    }

    // ---- butterfly argmin across the 16 lanes of each half-wave (first-index tie-break) ----
    #pragma unroll
    for (int j = 0; j < 8; ++j) {
        float d = minD[j]; int idx = minI[j];
        #pragma unroll
        for (int off = 8; off >= 1; off >>= 1) {
            float od = __shfl_xor(d, off, 32);
            int   oi = __shfl_xor(idx, off, 32);
            if (od < d || (od == d && oi < idx)) { d = od; idx = oi; }
        }
        minD[j] = d; minI[j] = idx;
    }

    // one-hot write + exact histogram (lanes 0 and 16 own 8 rows each)
    if (l16 == 0) {
        #pragma unroll
        for (int j = 0; j < 8; ++j) {
            int row = rowBase + j + 8 * half;
            int idx = minI[j];
            enc[(long long)row * NCODES + idx] = 1.0f;
            atomicAdd(&hist[idx], 1);
        }
    }

    // ---- quantized_st (== gathered codebook rows) + squared-error accumulation ----
    float lsum = 0.f;
    for (int r = 0; r < 16; ++r) {
        int j = r & 7, h = r >> 3;
        int idx = __shfl(minI[j], h * 16, 32);
        int row = rowBase + r;
        const float4* q4 = (const float4*)(cb + (long long)idx * ZDIM);
        const float4* x4 = (const float4*)(x + (long long)row * ZDIM);
        float* qo = quant + (long long)row * ZDIM;   // base offset 1 float -> scalar stores
        #pragma unroll
        for (int u = 0; u < 2; ++u) {
            int k4 = lane * 2 + u;
            float4 q = q4[k4], xi = x4[k4];
            float dx = q.x - xi.x, dy = q.y - xi.y, dz = q.z - xi.z, dw = q.w - xi.w;
            lsum += dx*dx + dy*dy + dz*dz + dw*dw;
            int k = k4 * 4;
            __builtin_nontemporal_store(q.x, &qo[k]);
            __builtin_nontemporal_store(q.y, &qo[k+1]);
            __builtin_nontemporal_store(q.z, &qo[k+2]);
            __builtin_nontemporal_store(q.w, &qo[k+3]);
        }
    }
    #pragma unroll
    for (int off = 16; off; off >>= 1) lsum += __shfl_xor(lsum, off, 32);
    if (lane == 0) atomicAdd(lossAcc, lsum);
}

__global__ void vq_finalize(const int* __restrict__ hist, const float* __restrict__ lossAcc,
                            float* __restrict__ loss_out, float* __restrict__ perp_out) {
    __shared__ float red[256];
    float s = 0.f;
    for (int c = threadIdx.x; c < NCODES; c += 256) {
        float p = (float)hist[c] * (1.0f / (float)N_ROWS);
        s += -p * logf(p + 1e-10f);
    }
    red[threadIdx.x] = s;
    __syncthreads();
    for (int w = 128; w; w >>= 1) {
        if (threadIdx.x < w) red[threadIdx.x] += red[threadIdx.x + w];
        __syncthreads();
    }
    if (threadIdx.x == 0) {
        *perp_out = expf(red[0]);
        *loss_out = 1.25f * (*lossAcc) / ((float)N_ROWS * (float)ZDIM);
    }
}

extern "C" void kernel_launch(void* const* d_in, const int* in_sizes, int n_in,
                              void* d_out, int out_size, void* d_ws, size_t ws_size,
                              hipStream_t stream) {
    (void)in_sizes; (void)n_in; (void)out_size; (void)ws_size;
    const float* x  = (const float*)d_in[0];
    const float* cb = (const float*)d_in[1];

    float* out   = (float*)d_out;
    float* loss  = out;                                        // [1]
    float* quant = out + 1;                                    // [131072*256]
    float* perp  = out + 1 + (long long)N_ROWS * ZDIM;         // [1]
    float* enc   = perp + 1;                                   // [131072*2048]

    float* norms   = (float*)d_ws;
    int*   hist    = (int*)((float*)d_ws + 2048);
    float* lossAcc = (float*)d_ws + 4096;

    const long long encF2 = (long long)N_ROWS * NCODES / 2;
    vq_zero    <<<65536, 256, 0, stream>>>(enc, encF2, (float*)d_ws);
    vq_cbnorm  <<<NCODES / 8, 256, 0, stream>>>(cb, norms);
    vq_main    <<<N_ROWS / ROWS_PER_BLOCK, 256, 0, stream>>>(x, cb, norms, hist, lossAcc, quant, enc);
    vq_finalize<<<1, 256, 0, stream>>>(hist, lossAcc, loss, perp);
}